// GINE_26783416058612
// MI455X (gfx1250) — compile-verified
//
#include <hip/hip_runtime.h>

#define NN 50000
#define EE 400000
#define DD 128

typedef __attribute__((ext_vector_type(2))) float v2f;
typedef __attribute__((ext_vector_type(8))) float v8f;
typedef __attribute__((ext_vector_type(4))) int v4i;

__device__ __forceinline__ float frelu(float v) { return v > 0.f ? v : 0.f; }

__device__ __forceinline__ void atomAddF(float* p, float v) {
  unsafeAtomicAdd(p, v);  // hardware global_atomic_add_f32
}

#if __has_builtin(__builtin_amdgcn_global_load_async_to_lds_b128)
#define HAVE_ASYNC_LDS 1
typedef __attribute__((address_space(1))) v4i GASv4i;
typedef __attribute__((address_space(3))) v4i LASv4i;
__device__ __forceinline__ void asyncWait0() {
#if __has_builtin(__builtin_amdgcn_s_wait_asynccnt)
  __builtin_amdgcn_s_wait_asynccnt(0);
#else
  asm volatile("s_wait_asynccnt 0x0" ::: "memory");
#endif
}
#else
#define HAVE_ASYNC_LDS 0
#endif

// ---------------------------------------------------------------------------
// Generic 16-row WMMA GEMM: C[M x NC] = A[M x KDIM] @ B[KDIM x NC] + bias
//                           (+ res)  (optional relu)
// Block = NC threads (NC/32 waves), each wave owns a 16x32 slab (2 tiles).
// A tile staged in LDS via async global->LDS copy (ASYNCcnt path); +4 float
// row padding keeps the l*LDA fragment reads bank-conflict free and every
// 16-byte chunk 16B-aligned (LDA*4 = 528 = 33*16).
// ---------------------------------------------------------------------------
template <int KDIM, int NC>
__global__ __launch_bounds__(NC) void gemm_wmma(
    const float* __restrict__ A, const float* __restrict__ B,
    const float* __restrict__ bias, const float* __restrict__ res,
    float* __restrict__ C, int do_relu) {
  constexpr int LDA = KDIM + 4;
  __shared__ float Alds[16 * LDA];
  const int tid = threadIdx.x;
  const long m0 = (long)blockIdx.x * 16;

  // stage 16 x KDIM tile of A in 16-byte chunks (contiguous source rows)
  const float* Asrc = A + m0 * KDIM;
#pragma unroll
  for (int i4 = tid; i4 < 16 * KDIM / 4; i4 += NC) {
    const int lin = i4 * 4;
    const int r = lin / KDIM, k = lin & (KDIM - 1);
#if HAVE_ASYNC_LDS
    __builtin_amdgcn_global_load_async_to_lds_b128(
        (GASv4i*)(Asrc + lin), (LASv4i*)&Alds[r * LDA + k], 0, 0);
#else
    const float4 v = *(const float4*)(Asrc + lin);
    float* dst = &Alds[r * LDA + k];
    dst[0] = v.x; dst[1] = v.y; dst[2] = v.z; dst[3] = v.w;
#endif
  }
#if HAVE_ASYNC_LDS
  asyncWait0();
#endif
  __syncthreads();

  const int wave = tid >> 5, lane = tid & 31;
  const int half = lane >> 4, l = lane & 15;
  const int n0 = wave * 32;
  v8f c0 = {}, c1 = {};
  const float* arow = &Alds[l * LDA];
  for (int k0 = 0; k0 < KDIM; k0 += 4) {
    const int ka = k0 + 2 * half;  // lanes 0-15: K=k0,k0+1 ; lanes 16-31: K=k0+2,k0+3
    v2f a, b0, b1;
    a.x = arow[ka]; a.y = arow[ka + 1];
    const float* bp = &B[(long)ka * NC + n0 + l];
    b0.x = bp[0];  b0.y = bp[NC];
    b1.x = bp[16]; b1.y = bp[16 + NC];
    c0 = __builtin_amdgcn_wmma_f32_16x16x4_f32(false, a, false, b0, (short)0, c0, false, false);
    c1 = __builtin_amdgcn_wmma_f32_16x16x4_f32(false, a, false, b1, (short)0, c1, false, false);
  }
  const float bi0 = bias[n0 + l], bi1 = bias[n0 + 16 + l];
#pragma unroll
  for (int r = 0; r < 8; ++r) {
    long grow = m0 + half * 8 + r;       // C/D layout: lanes16-31 hold rows M=r+8
    long o0 = grow * NC + n0 + l, o1 = o0 + 16;
    float v0 = c0[r] + bi0, v1 = c1[r] + bi1;
    if (res) { v0 += res[o0]; v1 += res[o1]; }
    if (do_relu) { v0 = frelu(v0); v1 = frelu(v1); }
    C[o0] = v0; C[o1] = v1;
  }
}

// ---------------------------------------------------------------------------
// Fused edge kernel: build conn tile (16 edges x 128) in LDS from gathered
// Qh/Kh rows, then WMMA against w2 with +b2 +poly_conn epilogue.
// Eh is E x 256 row-major: cols [0,128) = Qh, [128,256) = Kh.
// ---------------------------------------------------------------------------
__global__ __launch_bounds__(128) void edge_conn_gemm(
    const float* __restrict__ Eh, const int* __restrict__ dst,
    const int* __restrict__ src, const float* __restrict__ w2,
    const float* __restrict__ b2, const float* __restrict__ poly,
    float* __restrict__ conn2pre) {
  constexpr int K = 128, LDA = K + 4;
  __shared__ float lds[16 * LDA];
  __shared__ int idx_d[16];
  __shared__ int idx_s[16];
  const int f = threadIdx.x;
  const long e0 = (long)blockIdx.x * 16;
  if (f < 16)            idx_d[f]      = dst[e0 + f];
  else if (f < 32)       idx_s[f - 16] = src[e0 + f - 16];
  __syncthreads();
#pragma unroll 4
  for (int r = 0; r < 16; ++r) {
    const long e = e0 + r;
    const long d = idx_d[r], s = idx_s[r];
    const float qd = Eh[d * 256 + f];
    const float ks = Eh[s * 256 + 128 + f];
    const float qe = Eh[e * 256 + f];
    const float ke = Eh[e * 256 + 128 + f];
    lds[r * LDA + f] = frelu((qd + ks) * qe + ke);
  }
  __syncthreads();

  const int wave = threadIdx.x >> 5, lane = threadIdx.x & 31;
  const int half = lane >> 4, l = lane & 15;
  const int n0 = wave * 32;
  v8f c0 = {}, c1 = {};
  const float* arow = &lds[l * LDA];
  for (int k0 = 0; k0 < K; k0 += 4) {
    const int ka = k0 + 2 * half;
    v2f a, b0, b1;
    a.x = arow[ka]; a.y = arow[ka + 1];
    const float* bp = &w2[(long)ka * 128 + n0 + l];
    b0.x = bp[0];  b0.y = bp[128];
    b1.x = bp[16]; b1.y = bp[16 + 128];
    c0 = __builtin_amdgcn_wmma_f32_16x16x4_f32(false, a, false, b0, (short)0, c0, false, false);
    c1 = __builtin_amdgcn_wmma_f32_16x16x4_f32(false, a, false, b1, (short)0, c1, false, false);
  }
  const float bi0 = b2[n0 + l], bi1 = b2[n0 + 16 + l];
#pragma unroll
  for (int r = 0; r < 8; ++r) {
    long grow = e0 + half * 8 + r;
    long o0 = grow * 128 + n0 + l, o1 = o0 + 16;
    conn2pre[o0] = c0[r] + bi0 + poly[o0];
    conn2pre[o1] = c1[r] + bi1 + poly[o1];
  }
}

// per-column (feature) sum & sumsq of an M x 128 matrix -> stats[0..127]=sum,
// stats[128..255]=sumsq (must be pre-zeroed)
__global__ __launch_bounds__(128) void col_stats(const float* __restrict__ X,
                                                 int M, float* __restrict__ stats) {
  const int t = threadIdx.x;
  const long rpb = ((long)M + gridDim.x - 1) / gridDim.x;
  const long r0 = (long)blockIdx.x * rpb;
  const long r1 = (r0 + rpb < M) ? (r0 + rpb) : M;
  float s = 0.f, q = 0.f;
  for (long r = r0; r < r1; ++r) {
    float v = X[r * DD + t];
    s += v; q += v * v;
  }
  atomAddF(&stats[t], s);
  atomAddF(&stats[DD + t], q);
}

// stats[256..383]=mean, stats[384..511]=rsqrt(var+eps)   (biased var, /M)
__global__ __launch_bounds__(128) void stats_finalize(float* stats, float invM) {
  const int t = threadIdx.x;
  const float mean = stats[t] * invM;
  const float var = stats[DD + t] * invM - mean * mean;
  stats[2 * DD + t] = mean;
  stats[3 * DD + t] = rsqrtf(var + 1e-5f);
}

// conn2 = relu(bn(conn2)) in place, then atomic scatter-add into nh[dst]
__global__ __launch_bounds__(256) void bn_relu_scatter(
    float* __restrict__ conn2, const float* __restrict__ stats,
    const float* __restrict__ g, const float* __restrict__ b,
    const int* __restrict__ dst, float* __restrict__ nh) {
  const long idx = (long)blockIdx.x * 256 + threadIdx.x;
  const int f = idx & (DD - 1);
  const long e = idx >> 7;
  float v = (conn2[idx] - stats[2 * DD + f]) * stats[3 * DD + f] * g[f] + b[f];
  v = frelu(v);
  conn2[idx] = v;
  atomAddF(&nh[(long)dst[e] * DD + f], v);
}

// h_res = nh*dc0 + nh*sqrt_deg*dc1 + x
__global__ __launch_bounds__(256) void node_mix(
    const float* __restrict__ nh, const float* __restrict__ x,
    const float* __restrict__ sqrt_deg, const float* __restrict__ deg_coef,
    float* __restrict__ h_res) {
  const long idx = (long)blockIdx.x * 256 + threadIdx.x;
  const int f = idx & (DD - 1);
  const long n = idx >> 7;
  const float v = nh[idx];
  h_res[idx] = v * deg_coef[f * 2 + 0] + v * sqrt_deg[n] * deg_coef[f * 2 + 1] + x[idx];
}

__global__ __launch_bounds__(256) void bn_apply(
    const float* __restrict__ X, const float* __restrict__ stats,
    const float* __restrict__ g, const float* __restrict__ b,
    float* __restrict__ Y, int do_relu) {
  const long idx = (long)blockIdx.x * 256 + threadIdx.x;
  const int f = idx & (DD - 1);
  float v = (X[idx] - stats[2 * DD + f]) * stats[3 * DD + f] * g[f] + b[f];
  if (do_relu) v = frelu(v);
  Y[idx] = v;
}

extern "C" void kernel_launch(void* const* d_in, const int* in_sizes, int n_in,
                              void* d_out, int out_size, void* d_ws, size_t ws_size,
                              hipStream_t stream) {
  const float* x         = (const float*)d_in[0];
  const float* poly_conn = (const float*)d_in[1];
  const int*   poly_idx  = (const int*)d_in[2];
  const float* sqrt_deg  = (const float*)d_in[3];
  const float* w1        = (const float*)d_in[4];
  const float* b1        = (const float*)d_in[5];
  const float* w2        = (const float*)d_in[6];
  const float* b2        = (const float*)d_in[7];
  const float* conn_g    = (const float*)d_in[8];
  const float* conn_b    = (const float*)d_in[9];
  const float* deg_coef  = (const float*)d_in[10];
  const float* ffn_w1    = (const float*)d_in[11];
  const float* ffn_b1    = (const float*)d_in[12];
  const float* ffn_w2    = (const float*)d_in[13];
  const float* ffn_b2    = (const float*)d_in[14];
  const float* n1_g      = (const float*)d_in[15];
  const float* n1_b      = (const float*)d_in[16];
  const float* n2_g      = (const float*)d_in[17];
  const float* n2_b      = (const float*)d_in[18];

  const int* dst = poly_idx;        // poly_index[0]
  const int* src = poly_idx + EE;   // poly_index[1]

  float* ws = (float*)d_ws;
  float* Eh    = ws;                              // E*256
  float* nh    = ws + (size_t)EE * 256;           // N*128
  float* h_res = nh + (size_t)NN * DD;            // N*128
  float* stats = h_res + (size_t)NN * DD;         // 512
  // edge phase done -> reuse Eh region for node-side scratch
  float* a1    = ws;                              // N*128
  float* ffn_h = ws + (size_t)NN * DD;            // N*256
  float* pre2  = ws + (size_t)NN * DD * 3;        // N*128

  float* out_nh    = (float*)d_out;               // N*128
  float* out_conn2 = out_nh + (size_t)NN * DD;    // E*128

  // ---- edge pipeline ----
  (void)hipMemsetAsync(nh, 0, (size_t)NN * DD * sizeof(float), stream);
  gemm_wmma<128, 256><<<EE / 16, 256, 0, stream>>>(poly_conn, w1, b1, nullptr, Eh, 0);
  edge_conn_gemm<<<EE / 16, 128, 0, stream>>>(Eh, dst, src, w2, b2, poly_conn, out_conn2);
  (void)hipMemsetAsync(stats, 0, 512 * sizeof(float), stream);
  col_stats<<<512, 128, 0, stream>>>(out_conn2, EE, stats);
  stats_finalize<<<1, 128, 0, stream>>>(stats, 1.0f / EE);
  bn_relu_scatter<<<(EE * DD) / 256, 256, 0, stream>>>(out_conn2, stats, conn_g, conn_b, dst, nh);

  // ---- node pipeline ----
  node_mix<<<(NN * DD) / 256, 256, 0, stream>>>(nh, x, sqrt_deg, deg_coef, h_res);
  (void)hipMemsetAsync(stats, 0, 512 * sizeof(float), stream);
  col_stats<<<512, 128, 0, stream>>>(h_res, NN, stats);
  stats_finalize<<<1, 128, 0, stream>>>(stats, 1.0f / NN);
  bn_apply<<<(NN * DD) / 256, 256, 0, stream>>>(h_res, stats, n1_g, n1_b, a1, 0);

  gemm_wmma<128, 256><<<NN / 16, 256, 0, stream>>>(a1, ffn_w1, ffn_b1, nullptr, ffn_h, 1);
  gemm_wmma<256, 128><<<NN / 16, 128, 0, stream>>>(ffn_h, ffn_w2, ffn_b2, h_res, pre2, 0);

  (void)hipMemsetAsync(stats, 0, 512 * sizeof(float), stream);
  col_stats<<<512, 128, 0, stream>>>(pre2, NN, stats);
  stats_finalize<<<1, 128, 0, stream>>>(stats, 1.0f / NN);
  bn_apply<<<(NN * DD) / 256, 256, 0, stream>>>(pre2, stats, n2_g, n2_b, out_nh, 0);
}